// AELoss_13821204759295
// MI455X (gfx1250) — compile-verified
//
#include <hip/hip_runtime.h>

typedef float v2f __attribute__((ext_vector_type(2)));
typedef float v8f __attribute__((ext_vector_type(8)));

#define AE_B 64
#define AE_C 4
#define AE_H 256
#define AE_W 256
#define AE_N 128

__global__ void ae_zero_kernel(float* out) {
    if (threadIdx.x < 2) out[threadIdx.x] = 0.0f;
}

// One workgroup (256 threads = 8 wave32) per image.
// Wave w computes row-tile I=w of the 128x128 pairwise matrix (8 col-tiles).
__global__ __launch_bounds__(256) void ae_loss_kernel(
    const float* __restrict__ pred,
    const float* __restrict__ target,
    const int*   __restrict__ match,
    float* __restrict__ out)
{
    __shared__ float s_me[AE_N];   // s[n] = sum_c me[n][c]
    __shared__ float s_pull;
    __shared__ float s_push;

    const int b   = blockIdx.x;
    const int tid = threadIdx.x;

    if (tid == 0) { s_pull = 0.0f; s_push = 0.0f; }
    __syncthreads();

    // ---- Phase 1: gather keypoints, pull term, per-keypoint embedding sum ----
    if (tid < AE_N) {
        const int mbase = (b * AE_N + tid) * 4;   // match[b][n][2][2]
        const int tly = match[mbase + 0];
        const int tlx = match[mbase + 1];
        const int bry = match[mbase + 2];
        const int brx = match[mbase + 3];
        const int pbase = b * AE_C * AE_H * AE_W;
        const int tlo = tly * AE_W + tlx;
        const int bro = bry * AE_W + brx;
        float pull = 0.0f, ssum = 0.0f;
#pragma unroll
        for (int c = 0; c < AE_C; ++c) {
            const float tl = pred  [pbase + c * AE_H * AE_W + tlo];
            const float br = target[pbase + c * AE_H * AE_W + bro];
            const float d  = tl - br;            // tl-me = d/2, br-me = -d/2
            pull += 0.5f * d * d;                // (d/2)^2 + (d/2)^2
            ssum += 0.5f * (tl + br);            // me channel sum
        }
        s_me[tid] = ssum;
        atomicAdd(&s_pull, pull);                // ds_add_f32
    }
    __syncthreads();

    // ---- Phase 2: push term via WMMA rank-2 outer difference ----
    // conf_tile(16x16) = A(16x4) x B(4x16):  A = [s_i, 1, 0, 0], B = [1; -s_j; 0; 0]
    // A-layout (f32 16x4): lanes 0-15 hold K=0 (VGPR0) / K=1 (VGPR1); lanes 16-31 K=2/3 -> 0.
    const int lane = tid & 31;
    const int wave = tid >> 5;
    const float keep = (lane < 16) ? 1.0f : 0.0f;   // zero the K=2/3 half
    const int l16  = lane & 15;

    // A operand: loop-invariant (row-tile I == wave)
    const float sa = s_me[wave * 16 + l16];
    v2f a;
    a.x = sa * keep;   // K=0 row of A: s_i
    a.y = keep;        // K=1 row of A: 1

    float pushAcc = 0.0f;
#pragma unroll
    for (int t = 0; t < 8; ++t) {                   // col-tile J = t
        const float sb = s_me[t * 16 + l16];
        v2f bm;
        bm.x = keep;          // K=0 row of B: 1
        bm.y = -sb * keep;    // K=1 row of B: -s_j

        v8f acc = {};
        acc = __builtin_amdgcn_wmma_f32_16x16x4_f32(
            /*neg_a=*/false, a, /*neg_b=*/false, bm,
            /*c_mod=*/(short)0, acc, /*reuse_a=*/false, /*reuse_b=*/false);

        // Unmasked accumulate: diagonal elements are exactly 0 -> contribute
        // exactly relu(1-0)=1.0 each; the 128 spurious ones are subtracted
        // as an exact constant in the final step.
#pragma unroll
        for (int v = 0; v < 8; ++v) {
            pushAcc += fmaxf(0.0f, 1.0f - fabsf(acc[v]));   // relu(MARGIN - |conf|)
        }
    }
    atomicAdd(&s_push, pushAcc);                 // ds_add_f32
    __syncthreads();

    if (tid == 0) {
        atomicAdd(out + 0, 0.25f * s_pull * (1.0f / (float)AE_N));
        atomicAdd(out + 1, 0.25f * (s_push - (float)AE_N) *
                               (1.0f / (float)(AE_N * (AE_N - 1))));
    }
}

extern "C" void kernel_launch(void* const* d_in, const int* in_sizes, int n_in,
                              void* d_out, int out_size, void* d_ws, size_t ws_size,
                              hipStream_t stream) {
    const float* pred   = (const float*)d_in[0];
    const float* target = (const float*)d_in[1];
    const int*   match  = (const int*)d_in[2];
    float* out = (float*)d_out;

    hipLaunchKernelGGL(ae_zero_kernel, dim3(1), dim3(32), 0, stream, out);
    hipLaunchKernelGGL(ae_loss_kernel, dim3(AE_B), dim3(256), 0, stream,
                       pred, target, match, out);
}